// cnnToLSTMCustom_68478958568117
// MI455X (gfx1250) — compile-verified
//
#include <hip/hip_runtime.h>

// ---------------------------------------------------------------------------
// MI455X / gfx1250 implementation (round 2).
//
// Phase 0: transpose each weight tensor into B-fragment-friendly layout
//          wt[(ci*KSZ + k) * F_pad + f]  (zero-padded filter tail) so the
//          conv kernels' B loads are lane-coalesced instead of 32-line
//          gathers.
// Phase 1: four implicit-GEMM conv1d kernels, V_WMMA_F32_16X16X4_F32,
//          2x2 tile blocking per wave32 (32 rows x 32 filters), 4
//          independent accumulators -> 4 WMMAs per K-chunk with 8 coalesced
//          b32 loads. Results staged densely in d_ws.
// Phase 2: two pack kernels invert the scatter patterns analytically and
//          write each complex64 output element exactly once (float2 / b64
//          stores; 223.5 MB output traffic dominates => memory bound).
// ---------------------------------------------------------------------------

typedef float v2f __attribute__((ext_vector_type(2)));
typedef float v8f __attribute__((ext_vector_type(8)));

#define N_BATCH 16
#define C_IN    256
#define L_IN    2048

// ------------------- weight transpose: w(F,C,K) -> wt(C*K, F_pad) ----------
__global__ __launch_bounds__(256)
void transpose_w_kernel(const float* __restrict__ w, float* __restrict__ wt,
                        int KSZ, int F_used, int F_pad)
{
    const int idx = blockIdx.x * blockDim.x + threadIdx.x;
    const int total = C_IN * KSZ * F_pad;
    if (idx >= total) return;
    const int f    = idx % F_pad;
    const int kidx = idx / F_pad;          // ci*KSZ + k
    const int ci   = kidx / KSZ;
    const int k    = kidx - ci * KSZ;
    wt[idx] = (f < F_used) ? w[((size_t)f * C_IN + ci) * KSZ + k] : 0.f;
}

// ------------------- implicit-GEMM conv1d with f32 WMMA --------------------
template<int KSZ, int STRIDE, int PAD>
__global__ __launch_bounds__(32)
void conv_wmma_kernel(const float* __restrict__ x,
                      const float* __restrict__ wt,   // (C*KSZ, F_pad), padded
                      const float* __restrict__ bias,
                      float* __restrict__ t,
                      int S_out, int F_used, int F_pad)
{
    const int lane = threadIdx.x;   // wave32
    const int l = lane & 15;
    const int h = lane >> 4;        // 0 or 1

    const int f0 = blockIdx.x * 32; // filter super-tile (2 x 16)
    const int s0 = blockIdx.y * 32; // row super-tile    (2 x 16)
    const int n  = blockIdx.z;

    const float* __restrict__ xn = x + (size_t)n * C_IN * L_IN;

    const int sA0 = s0 + l;          // A-fragment rows (M = l) for M-tile 0/1
    const int sA1 = s0 + 16 + l;
    const int fB0 = f0 + l;          // B-fragment cols (N = l) for N-tile 0/1
    const int fB1 = f0 + 16 + l;

    v8f acc00 = {}, acc01 = {}, acc10 = {}, acc11 = {};

    const int Ktot = C_IN * KSZ;     // multiple of 4
    for (int kk = 0; kk < Ktot; kk += 4) {
        // This lane supplies flattened-K indices kA = kk+2h and kA+1.
        const int kA  = kk + 2 * h;
        const int ci0 = kA / KSZ,       k0 = kA - ci0 * KSZ;
        const int ci1 = (kA + 1) / KSZ, k1 = (kA + 1) - ci1 * KSZ;

        // A[M=s][K] = x[n][ci][s*STRIDE + k - PAD]  (zero outside [0, L))
        const int p00 = sA0 * STRIDE + k0 - PAD;
        const int p01 = sA0 * STRIDE + k1 - PAD;
        const int p10 = sA1 * STRIDE + k0 - PAD;
        const int p11 = sA1 * STRIDE + k1 - PAD;
        v2f a0, a1;
        a0[0] = (p00 >= 0 && p00 < L_IN) ? xn[ci0 * L_IN + p00] : 0.f;
        a0[1] = (p01 >= 0 && p01 < L_IN) ? xn[ci1 * L_IN + p01] : 0.f;
        a1[0] = (p10 >= 0 && p10 < L_IN) ? xn[ci0 * L_IN + p10] : 0.f;
        a1[1] = (p11 >= 0 && p11 < L_IN) ? xn[ci1 * L_IN + p11] : 0.f;

        // B[K][N=f] = wt[K * F_pad + f]  (coalesced across lanes)
        const float* __restrict__ wr0 = wt + (size_t)kA * F_pad;
        const float* __restrict__ wr1 = wr0 + F_pad;
        v2f b0, b1;
        b0[0] = wr0[fB0];  b0[1] = wr1[fB0];
        b1[0] = wr0[fB1];  b1[1] = wr1[fB1];

        // 4 independent accumulators -> pipelined matrix ops
        acc00 = __builtin_amdgcn_wmma_f32_16x16x4_f32(false, a0, false, b0,
                                                      (short)0, acc00, false, false);
        acc01 = __builtin_amdgcn_wmma_f32_16x16x4_f32(false, a0, false, b1,
                                                      (short)0, acc01, false, false);
        acc10 = __builtin_amdgcn_wmma_f32_16x16x4_f32(false, a1, false, b0,
                                                      (short)0, acc10, false, false);
        acc11 = __builtin_amdgcn_wmma_f32_16x16x4_f32(false, a1, false, b1,
                                                      (short)0, acc11, false, false);
    }

    // D layout: VGPR j of lane (l + 16h) holds D[M = j + 8h][N = l]
    const float bb0 = (fB0 < F_used) ? bias[fB0] : 0.f;
    const float bb1 = (fB1 < F_used) ? bias[fB1] : 0.f;

    #pragma unroll
    for (int j = 0; j < 8; ++j) {
        const int r0 = s0 + j + 8 * h;
        const int r1 = r0 + 16;
        if (r0 < S_out) {
            float* row = t + ((size_t)n * S_out + r0) * F_pad;
            if (fB0 < F_used) row[fB0] = acc00[j] + bb0;
            if (fB1 < F_used) row[fB1] = acc01[j] + bb1;
        }
        if (r1 < S_out) {
            float* row = t + ((size_t)n * S_out + r1) * F_pad;
            if (fB0 < F_used) row[fB0] = acc10[j] + bb0;
            if (fB1 < F_used) row[fB1] = acc11[j] + bb1;
        }
    }
}

// ------------------- pack: upper = [complex(s2,0) | complex(0,s4)] ----------
__global__ __launch_bounds__(256)
void pack_upper_kernel(const float* __restrict__ t2,   // (16*1023, 256)
                       const float* __restrict__ t4,   // (16*1023, 128)
                       float2* __restrict__ out)       // (16*1023, 1024) complex
{
    const int idx = blockIdx.x * blockDim.x + threadIdx.x;
    const int total = N_BATCH * 1023 * 1024;
    if (idx >= total) return;
    const int c   = idx & 1023;
    const int row = idx >> 10;          // n*1023 + s

    float re = 0.f, im = 0.f;
    if (c < 512) {
        // s2: positions 2a+1, 2a+2 <- channel a (a in [0,255))
        const int p   = c;
        const int fid = (p & 1) ? ((p - 1) >> 1) : ((p >> 1) - 1);
        if (fid >= 0 && fid < 255)
            re = t2[(size_t)row * 256 + fid];
    } else {
        // s4: positions 4a+{1,3,4,6} <- channel a (a in [0,127))
        const int p = c - 512;
        const int r = p & 3;
        int fid;
        if      (r == 1) fid = (p - 1) >> 2;
        else if (r == 3) fid = (p - 3) >> 2;
        else if (r == 0) fid = (p >> 2) - 1;
        else             fid = (p - 6) >> 2;   // r == 2 (negative -> invalid)
        if (fid >= 0 && fid < 127)
            im = t4[(size_t)row * 128 + fid];
    }
    out[idx] = make_float2(re, im);
}

// ------------------- pack: middle = [complex(s3,0) | complex(0,s6)] ---------
__global__ __launch_bounds__(256)
void pack_middle_kernel(const float* __restrict__ t3,  // (16*682, 96) : 86 used
                        const float* __restrict__ t6,  // (16*682, 96) : 85 used
                        float2* __restrict__ out)      // (16*682, 1024) complex
{
    const int idx = blockIdx.x * blockDim.x + threadIdx.x;
    const int total = N_BATCH * 682 * 1024;
    if (idx >= total) return;
    const int c   = idx & 1023;
    const int row = idx >> 10;          // n*682 + s

    float re = 0.f, im = 0.f;
    if (c < 512) {
        // s3: every odd p; fid = i for p in {6i-1, 6i+1, 6i+3}, fid in [0,86)
        const int p = c;
        if (p & 1) {
            const int r = p % 6;        // 1, 3 or 5
            const int fid = (r == 5) ? (p + 1) / 6
                          : (r == 1) ? (p - 1) / 6
                                     : (p - 3) / 6;
            if (fid < 86)
                re = t3[(size_t)row * 96 + fid];
        }
    } else {
        // s6: p in {6i-1,6i+1,6i+3,6i+4,6i+6,6i+8}; all residues mod 6; fid in [0,85)
        const int p = c - 512;
        const int r = p % 6;
        int fid;
        switch (r) {
            case 0:  fid = p / 6 - 1;   break;
            case 1:  fid = (p - 1) / 6; break;
            case 2:  fid = (p - 8) / 6; break;  // negative -> invalid
            case 3:  fid = (p - 3) / 6; break;
            case 4:  fid = (p - 4) / 6; break;
            default: fid = (p + 1) / 6; break;  // r == 5
        }
        if (fid >= 0 && fid < 85)
            im = t6[(size_t)row * 96 + fid];
    }
    out[idx] = make_float2(re, im);
}

extern "C" void kernel_launch(void* const* d_in, const int* in_sizes, int n_in,
                              void* d_out, int out_size, void* d_ws, size_t ws_size,
                              hipStream_t stream)
{
    (void)in_sizes; (void)n_in; (void)out_size; (void)ws_size;

    const float* x  = (const float*)d_in[0];
    const float* w2 = (const float*)d_in[1];
    const float* b2 = (const float*)d_in[2];
    const float* w4 = (const float*)d_in[3];
    const float* b4 = (const float*)d_in[4];
    const float* w3 = (const float*)d_in[5];
    const float* b3 = (const float*)d_in[6];
    const float* w6 = (const float*)d_in[7];
    const float* b6 = (const float*)d_in[8];

    // scratch layout (f32): ~35.5 MB total
    float* t2  = (float*)d_ws;                                 // 16*1023*256
    float* t4  = t2  + (size_t)N_BATCH * 1023 * 256;           // 16*1023*128
    float* t3  = t4  + (size_t)N_BATCH * 1023 * 128;           // 16*682*96
    float* t6  = t3  + (size_t)N_BATCH * 682 * 96;             // 16*682*96
    float* wt2 = t6  + (size_t)N_BATCH * 682 * 96;             // 512*256
    float* wt4 = wt2 + (size_t)C_IN * 2 * 256;                 // 1024*128
    float* wt3 = wt4 + (size_t)C_IN * 4 * 128;                 // 768*96
    float* wt6 = wt3 + (size_t)C_IN * 3 * 96;                  // 1536*96

    // Phase 0: weight transposes (tiny)
    transpose_w_kernel<<<(C_IN * 2 * 256 + 255) / 256, 256, 0, stream>>>(w2, wt2, 2, 255, 256);
    transpose_w_kernel<<<(C_IN * 4 * 128 + 255) / 256, 256, 0, stream>>>(w4, wt4, 4, 127, 128);
    transpose_w_kernel<<<(C_IN * 3 *  96 + 255) / 256, 256, 0, stream>>>(w3, wt3, 3,  86,  96);
    transpose_w_kernel<<<(C_IN * 6 *  96 + 255) / 256, 256, 0, stream>>>(w6, wt6, 6,  85,  96);

    // Phase 1: WMMA convolutions (grid = (f_tiles32, s_tiles32, N), 1 wave/block)
    conv_wmma_kernel<2, 1, 0><<<dim3(8, 32, N_BATCH), 32, 0, stream>>>(
        x, wt2, b2, t2, 1023, 255, 256);
    conv_wmma_kernel<4, 2, 0><<<dim3(4, 32, N_BATCH), 32, 0, stream>>>(
        x, wt4, b4, t4, 1023, 127, 128);
    conv_wmma_kernel<3, 3, 1><<<dim3(3, 22, N_BATCH), 32, 0, stream>>>(
        x, wt3, b3, t3,  682,  86,  96);
    conv_wmma_kernel<6, 3, 1><<<dim3(3, 22, N_BATCH), 32, 0, stream>>>(
        x, wt6, b6, t6,  682,  85,  96);

    // Phase 2: pack complex outputs (each element written exactly once)
    float2* out_upper  = (float2*)d_out;
    float2* out_middle = out_upper + (size_t)N_BATCH * 1023 * 1024;

    const int totU = N_BATCH * 1023 * 1024;
    const int totM = N_BATCH *  682 * 1024;
    pack_upper_kernel <<<(totU + 255) / 256, 256, 0, stream>>>(t2, t4, out_upper);
    pack_middle_kernel<<<(totM + 255) / 256, 256, 0, stream>>>(t3, t6, out_middle);
}